// QuantumAttention_65481071406371
// MI455X (gfx1250) — compile-verified
//
#include <hip/hip_runtime.h>

// Quantum attention collapses exactly: RZ phase cancels in |state|^2, so
// q_q = k_q = v_q = const vector z(params); softmax of a constant matrix is
// uniform; out rows all equal y = w_mix @ c with c[f] = z[f & 3].
// Kernel 1 (1 block): compute z, then y = w_mix * c via v_wmma_f32_16x16x4_f32.
// Kernel 2: broadcast y (128 floats) across the 8x1024x128 output (4 MB,
// pure store-bandwidth bound: ~0.17 us at 23.3 TB/s).

typedef __attribute__((ext_vector_type(2))) float v2f;
typedef __attribute__((ext_vector_type(8))) float v8f;

__global__ __launch_bounds__(256) void qa_zy_wmma(const float* __restrict__ w_mix,
                                                  const float* __restrict__ params,
                                                  float* __restrict__ y /*128 floats*/) {
    const int lane = threadIdx.x & 31;
    const int wave = threadIdx.x >> 5;   // 8 waves, wave m owns rows [16m,16m+16)

    // ---- z[w] = sum_b base[b]^2 * (1 - 2*bit_w(f(b))), f = CNOT chain ----
    const float h0 = 0.5f * params[0], h1 = 0.5f * params[1];
    const float h2 = 0.5f * params[2], h3 = 0.5f * params[3];
    const float q0[2] = {__builtin_cosf(h0), __builtin_sinf(h0)};
    const float q1[2] = {__builtin_cosf(h1), __builtin_sinf(h1)};
    const float q2[2] = {__builtin_cosf(h2), __builtin_sinf(h2)};
    const float q3[2] = {__builtin_cosf(h3), __builtin_sinf(h3)};
    float z0 = 0.f, z1 = 0.f, z2 = 0.f, z3 = 0.f;
#pragma unroll
    for (int b = 0; b < 16; ++b) {
        const int b0 = (b >> 3) & 1, b1 = (b >> 2) & 1, b2 = (b >> 1) & 1, b3 = b & 1;
        const float amp = q0[b0] * q1[b1] * q2[b2] * q3[b3];
        const float p = amp * amp;
        const int n1 = b1 ^ b0;          // CNOT(0,1)
        const int n2 = b2 ^ n1;          // CNOT(1,2)
        const int n3 = b3 ^ n2;          // CNOT(2,3)
        const int n0 = b0 ^ n3;          // CNOT(3,0)
        z0 += p * (1.0f - 2.0f * (float)n0);
        z1 += p * (1.0f - 2.0f * (float)n1);
        z2 += p * (1.0f - 2.0f * (float)n2);
        z3 += p * (1.0f - 2.0f * (float)n3);
    }

    // ---- y = w_mix * c, c[f] = z[f&3], via V_WMMA_F32_16X16X4_F32 ----
    // B(4x16): every column = (z0,z1,z2,z3); layout mirrors A 16x4:
    //   lanes 0-15 hold K=0,1 (vgpr0,1), lanes 16-31 hold K=2,3.
    v2f bmat;
    bmat.x = (lane < 16) ? z0 : z2;
    bmat.y = (lane < 16) ? z1 : z3;

    // A(16x4) tile of w_mix rows [16*wave .. 16*wave+15]:
    //   lane L<16 -> row L, K=k+0,k+1 ; lane L>=16 -> row L-16, K=k+2,k+3
    const int row  = wave * 16 + (lane & 15);
    const int koff = (lane >> 4) * 2;
    const float* wrow = w_mix + row * 128 + koff;

    v8f acc = {};
#pragma unroll
    for (int k = 0; k < 128; k += 4) {
        v2f a;
        a.x = wrow[k + 0];
        a.y = wrow[k + 1];
        // 8 args: (neg_a, A, neg_b, B, c_mod, C, reuse_a, reuse_b)
        acc = __builtin_amdgcn_wmma_f32_16x16x4_f32(
            false, a, false, bmat, (short)0, acc, false, false);
    }

    // D layout: vgpr r holds M=r (lanes 0-15, N=lane) / M=r+8 (lanes 16-31).
    // All N columns are identical; lanes with (lane&15)==0 write their 8 rows.
    if ((lane & 15) == 0) {
        const int rbase = wave * 16 + (lane >> 4) * 8;
#pragma unroll
        for (int r = 0; r < 8; ++r) y[rbase + r] = acc[r];
    }
}

__global__ __launch_bounds__(256) void qa_broadcast(const float* __restrict__ y,
                                                    float* __restrict__ out, int n4) {
    const int i = blockIdx.x * blockDim.x + threadIdx.x;
    if (i < n4) {
        const float4* y4 = (const float4*)y;     // 32 float4 groups, stays in cache
        ((float4*)out)[i] = y4[i & 31];          // e-range repeats every 128 floats
    }
}

extern "C" void kernel_launch(void* const* d_in, const int* in_sizes, int n_in,
                              void* d_out, int out_size, void* d_ws, size_t ws_size,
                              hipStream_t stream) {
    // inputs: 0=x, 1=w_q, 2=w_k, 3=w_v, 4=w_mix (128x128), 5=params (4)
    const float* w_mix  = (const float*)d_in[4];
    const float* params = (const float*)d_in[5];
    float* y   = (float*)d_ws;      // 128 floats of scratch
    float* out = (float*)d_out;     // 8*1024*128 floats

    qa_zy_wmma<<<1, 256, 0, stream>>>(w_mix, params, y);

    const int n4 = out_size / 4;    // 262144 float4 stores
    qa_broadcast<<<(n4 + 255) / 256, 256, 0, stream>>>(y, out, n4);
}